// SeparableCrossAttention_55061480734998
// MI455X (gfx1250) — compile-verified
//
#include <hip/hip_runtime.h>
#include <hip/hip_bf16.h>
#include <math.h>

typedef __attribute__((ext_vector_type(2))) float v2f;
typedef __attribute__((ext_vector_type(8))) float v8f;

// Problem constants
#define BB 4
#define CC 128
#define SS 256   // N == M == 256
#define FF 256
#define HH 8
#define DD 32
#define HID 256
#define NMEM 4
#define SCALE 0.17677669529663687f  // 1/sqrt(32)

// -------------------------------------------------------------------------
// Generic strided + batched fp32 WMMA GEMM:  O[m,n] = sum_k A[m,k]*B[k,n]
// One 16x16 output tile per wave, K stepped by 4 (V_WMMA_F32_16X16X4_F32).
// A 16x4 layout: lanes 0-15 hold row=lane, k={0,1}; lanes 16-31 row=lane-16,
// k={2,3}. B 4x16: lanes 0-15 col=lane k={0,1}; lanes 16-31 col, k={2,3}.
// C/D: VGPR i -> (m = i + (lane>>4)*8, n = lane&15).
// -------------------------------------------------------------------------
__global__ __launch_bounds__(256) void wmma_gemm(
    const float* __restrict__ A, const float* __restrict__ B,
    float* __restrict__ O, int M, int N, int K,
    int saM, int saK, int sbK, int sbN, int soM, int soN,
    long baA, long baB, long baO)
{
  int batch = blockIdx.y;
  A += (long)batch * baA;
  B += (long)batch * baB;
  O += (long)batch * baO;
  int wave = threadIdx.x >> 5, lane = threadIdx.x & 31;
  int tilesN = N >> 4;
  int tiles  = (M >> 4) * tilesN;
  int tile = blockIdx.x * 8 + wave;
  if (tile >= tiles) return;                 // wave-uniform exit
  int tm = (tile / tilesN) << 4;
  int tn = (tile % tilesN) << 4;
  int r = lane & 15, half = lane >> 4;
  v8f acc = {};
  for (int k0 = 0; k0 < K; k0 += 4) {
    int ka = k0 + half * 2;
    v2f a, b;
    a.x = A[(long)(tm + r) * saM + (long)ka * saK];
    a.y = A[(long)(tm + r) * saM + (long)(ka + 1) * saK];
    b.x = B[(long)ka * sbK + (long)(tn + r) * sbN];
    b.y = B[(long)(ka + 1) * sbK + (long)(tn + r) * sbN];
    acc = __builtin_amdgcn_wmma_f32_16x16x4_f32(false, a, false, b,
                                                (short)0, acc, false, false);
  }
  int om = tm + half * 8, on = tn + r;
#pragma unroll
  for (int i = 0; i < 8; ++i)
    O[(long)(om + i) * soM + (long)on * soN] = acc[i];
}

// -------------------------------------------------------------------------
// Fused scores + softmax. Block = (b, h, 16 s-rows), 8 waves.
// Each wave: two 16x16 WMMA score tiles (t = wave*32 .. +32), K=32 over D.
// Softmax over [4 mem scores | 256 kv scores] per row via wave shuffles.
// P==0 writes ATTN[b,h,s,t]; P==1 writes transposed ATTN[b,h,t,s].
// -------------------------------------------------------------------------
template <int P>
__global__ __launch_bounds__(256) void scores_softmax(
    const float* __restrict__ Q, const float* __restrict__ Km,
    const float* __restrict__ memk,
    float* __restrict__ ATTN, float* __restrict__ AMEM)
{
  __shared__ float sc[16][264];
  int bid = blockIdx.x;              // b*H*(S/16) = 512
  int s0 = (bid & 15) << 4;
  int h  = (bid >> 4) & 7;
  int b  = bid >> 7;
  int wave = threadIdx.x >> 5, lane = threadIdx.x & 31;
  int r = lane & 15, half = lane >> 4;

  // mem scores (cols 256..259) and -inf pad (260..263)
  for (int idx = threadIdx.x; idx < 16 * 8; idx += 256) {
    int s = idx >> 3, j = idx & 7;
    if (j < 4) {
      const float* qp = Q + ((long)(b * SS + s0 + s)) * HID + h * DD;
      const float* mk = memk + ((long)h * NMEM + j) * DD;
      float acc = 0.f;
      for (int d = 0; d < DD; ++d) acc += qp[d] * mk[d];
      sc[s][256 + j] = acc * SCALE;
    } else {
      sc[s][256 + j] = -INFINITY;
    }
  }

  // kv scores via WMMA: A = Q tile (16s x 4d), B = K^T tile (4d x 16t)
  const float* qbase = Q + ((long)b * SS + s0) * HID + h * DD;
  for (int sub = 0; sub < 2; ++sub) {
    int t0 = wave * 32 + sub * 16;
    const float* kbase = Km + ((long)b * SS + t0) * HID + h * DD;
    v8f acc = {};
    for (int k0 = 0; k0 < DD; k0 += 4) {
      int ka = k0 + half * 2;
      v2f a, bb;
      a.x  = qbase[(long)r * HID + ka];
      a.y  = qbase[(long)r * HID + ka + 1];
      bb.x = kbase[(long)r * HID + ka];
      bb.y = kbase[(long)r * HID + ka + 1];
      acc = __builtin_amdgcn_wmma_f32_16x16x4_f32(false, a, false, bb,
                                                  (short)0, acc, false, false);
    }
#pragma unroll
    for (int i = 0; i < 8; ++i)
      sc[half * 8 + i][t0 + r] = acc[i] * SCALE;
  }
  __syncthreads();

  // softmax: wave handles rows 2*wave, 2*wave+1; lane-parallel over 264 cols
  for (int rr = 0; rr < 2; ++rr) {
    int s = wave * 2 + rr;
    float mx = -INFINITY;
    for (int j = lane; j < 264; j += 32) mx = fmaxf(mx, sc[s][j]);
    for (int off = 16; off; off >>= 1) mx = fmaxf(mx, __shfl_xor(mx, off, 32));
    float sum = 0.f;
    for (int j = lane; j < 264; j += 32) {
      float e = __expf(sc[s][j] - mx);
      sc[s][j] = e;
      sum += e;
    }
    for (int off = 16; off; off >>= 1) sum += __shfl_xor(sum, off, 32);
    float inv = 1.f / sum;
    for (int j = lane; j < 264; j += 32) sc[s][j] *= inv;
  }
  __syncthreads();

  long abase = (long)(b * HH + h);
  if (P == 0) {
    int t = threadIdx.x;                       // coalesced over t
    for (int s = 0; s < 16; ++s)
      ATTN[(abase * SS + s0 + s) * SS + t] = sc[s][t];
  } else {
    int s = threadIdx.x & 15;                  // transposed, coalesced over s
    int tb = threadIdx.x >> 4;
    for (int i = 0; i < 16; ++i) {
      int t = tb + 16 * i;
      ATTN[(abase * SS + t) * SS + s0 + s] = sc[s][t];
    }
  }
  if (threadIdx.x < 64) {
    int s = threadIdx.x >> 2, j = threadIdx.x & 3;
    AMEM[(abase * SS + s0 + s) * NMEM + j] = sc[s][256 + j];
  }
}

// -------------------------------------------------------------------------
// Row pass y: block per (b,s). Attention slice (8h x 256t) cached in LDS,
// waves stripe over c, lanes over t (x reads fully coalesced, x read once).
// y[b,s,h,c] = sum_t a[h,t] * x[b,c,s,t]
// -------------------------------------------------------------------------
__global__ __launch_bounds__(256) void y_row(
    const float* __restrict__ x, const float* __restrict__ ATTN,
    float* __restrict__ Y)
{
  __shared__ float a[8][256];
  int b = blockIdx.x >> 8, s = blockIdx.x & 255;
  for (int i = threadIdx.x; i < 8 * 256; i += 256) {
    int h = i >> 8, t = i & 255;
    a[h][t] = ATTN[(((long)b * HH + h) * SS + s) * SS + t];
  }
  __syncthreads();
  int wave = threadIdx.x >> 5, lane = threadIdx.x & 31;
  for (int ci = 0; ci < 16; ++ci) {
    int c = wave + 8 * ci;
    const float* xp = x + (((long)b * CC + c) * SS + s) * SS;
    float acc[8] = {};
    for (int it = 0; it < 8; ++it) {
      int t = lane + 32 * it;
      float xv = xp[t];
#pragma unroll
      for (int h = 0; h < 8; ++h) acc[h] += a[h][t] * xv;
    }
#pragma unroll
    for (int h = 0; h < 8; ++h)
      for (int off = 16; off; off >>= 1)
        acc[h] += __shfl_xor(acc[h], off, 32);
    if (lane == 0) {
      float* yp = Y + (((long)b * SS + s) * HH) * CC + c;
#pragma unroll
      for (int h = 0; h < 8; ++h) yp[(long)h * CC] = acc[h];
    }
  }
}

// -------------------------------------------------------------------------
// Col pass y: y[b,s,h,c] = sum_t ATTNT[b,h,t,s] * x[b,c,t,s].
// Block = (b, 32-s group, 8-c group); lanes over s (both streams coalesced),
// waves split the t range, cross-wave reduce via LDS ds_add_f32.
// x is still read exactly once in total.
// -------------------------------------------------------------------------
__global__ __launch_bounds__(256) void y_col(
    const float* __restrict__ x, const float* __restrict__ ATTNT,
    float* __restrict__ Y)
{
  __shared__ float ly[32][64];                 // [s][h*8+c]
  int cg = blockIdx.x & 15;
  int sg = (blockIdx.x >> 4) & 7;
  int b  = blockIdx.x >> 7;
  int s0 = sg * 32, c0 = cg * 8;
  for (int i = threadIdx.x; i < 32 * 64; i += 256) ly[i >> 6][i & 63] = 0.f;
  __syncthreads();
  int wave = threadIdx.x >> 5, lane = threadIdx.x & 31;
  int s = s0 + lane;
  float acc[8][8] = {};
  for (int tt = 0; tt < 32; ++tt) {
    int t = wave * 32 + tt;
    float av[8];
#pragma unroll
    for (int h = 0; h < 8; ++h)
      av[h] = ATTNT[(((long)b * HH + h) * SS + t) * SS + s];
#pragma unroll
    for (int c = 0; c < 8; ++c) {
      float xv = x[(((long)b * CC + c0 + c) * SS + t) * SS + s];
#pragma unroll
      for (int h = 0; h < 8; ++h) acc[h][c] += av[h] * xv;
    }
  }
#pragma unroll
  for (int h = 0; h < 8; ++h)
#pragma unroll
    for (int c = 0; c < 8; ++c)
      atomicAdd(&ly[lane][h * 8 + c], acc[h][c]);
  __syncthreads();
  for (int i = threadIdx.x; i < 32 * 64; i += 256) {
    int si = i >> 6, h = (i & 63) >> 3, c = i & 7;
    Y[(((long)b * SS + s0 + si) * HH + h) * CC + c0 + c] = ly[si][i & 63];
  }
}

// O1[b,s,h*32+d] += sum_k AMEM[b,h,s,k] * mem_v[h,k,d]
__global__ __launch_bounds__(256) void mem_add(
    const float* __restrict__ AMEM, const float* __restrict__ memv,
    float* __restrict__ O1)
{
  int bs = blockIdx.x;
  int e = threadIdx.x, h = e >> 5, d = e & 31;
  int b = bs >> 8, s = bs & 255;
  const float* am = AMEM + (((long)b * HH + h) * SS + s) * NMEM;
  float acc = 0.f;
#pragma unroll
  for (int k = 0; k < NMEM; ++k)
    acc += am[k] * memv[((long)h * NMEM + k) * DD + d];
  O1[(long)bs * HID + e] += acc;
}

// out[b,o,n,m] = PR[b,o,n] + PC[b,o,m]   (the only 134MB write)
__global__ __launch_bounds__(256) void final_add(
    const float* __restrict__ PR, const float* __restrict__ PC,
    float* __restrict__ out)
{
  long bon = blockIdx.x;                       // b*C*N + o*N + n
  int n = (int)(bon & 255);
  long bo = bon >> 8;
  float pr = PR[bo * SS + n];
  const float* pc = PC + bo * SS;
  float* op = out + bon * SS;
  op[threadIdx.x] = pr + pc[threadIdx.x];
}

// -------------------------------------------------------------------------
// Workspace layout (floats)
// -------------------------------------------------------------------------
#define OFF_Q    0L
#define OFF_K    262144L
#define OFF_ATTN 524288L      // 4*8*256*256
#define OFF_AMEM 2621440L
#define OFF_Y    2654208L     // 4*256*8*128
#define OFF_O1   3702784L
#define OFF_AX0  3964928L
#define OFF_AX1  4096000L
#define OFF_PR   4227072L
#define OFF_PC   4358144L     // total 4489216 floats = ~18 MB

extern "C" void kernel_launch(void* const* d_in, const int* in_sizes, int n_in,
                              void* d_out, int out_size, void* d_ws, size_t ws_size,
                              hipStream_t stream) {
  const float* x      = (const float*)d_in[0];
  const float* row_ft = (const float*)d_in[1];
  const float* col_ft = (const float*)d_in[2];
  const float* Wq     = (const float*)d_in[3];
  const float* Wk     = (const float*)d_in[4];
  const float* Wv     = (const float*)d_in[5];
  const float* Wo     = (const float*)d_in[6];
  const float* mem_kv = (const float*)d_in[7];  // [2,H,NMEM,D]
  const float* Wr     = (const float*)d_in[8];  // [C, 2C]
  float* out = (float*)d_out;
  float* ws  = (float*)d_ws;

  float* Qw   = ws + OFF_Q;
  float* Kw   = ws + OFF_K;
  float* ATTN = ws + OFF_ATTN;
  float* AMEM = ws + OFF_AMEM;
  float* Yw   = ws + OFF_Y;
  float* O1   = ws + OFF_O1;
  float* AX[2] = { ws + OFF_AX0, ws + OFF_AX1 };
  float* PRJ[2] = { ws + OFF_PR, ws + OFF_PC };
  const float* mem_k = mem_kv;
  const float* mem_v = mem_kv + (long)HH * NMEM * DD;

  const int BS = BB * SS;  // 1024

  for (int p = 0; p < 2; ++p) {
    const float* qft = p ? col_ft : row_ft;
    const float* kft = p ? row_ft : col_ft;

    // Q = qft @ Wq   (1024x256x256)
    wmma_gemm<<<dim3(((BS/16)*(HID/16) + 7)/8, 1), 256, 0, stream>>>(
        qft, Wq, Qw, BS, HID, FF, FF, 1, HID, 1, HID, 1, 0, 0, 0);
    // K = kft @ Wk
    wmma_gemm<<<dim3(((BS/16)*(HID/16) + 7)/8, 1), 256, 0, stream>>>(
        kft, Wk, Kw, BS, HID, FF, FF, 1, HID, 1, HID, 1, 0, 0, 0);

    // fused scores + softmax (transposed attention layout for col pass)
    if (p == 0)
      scores_softmax<0><<<dim3(BB*HH*(SS/16)), 256, 0, stream>>>(Qw, Kw, mem_k, ATTN, AMEM);
    else
      scores_softmax<1><<<dim3(BB*HH*(SS/16)), 256, 0, stream>>>(Qw, Kw, mem_k, ATTN, AMEM);

    // y = attention-weighted sum over x (the bandwidth-dominant stage)
    if (p == 0)
      y_row<<<dim3(BB*SS), 256, 0, stream>>>(x, ATTN, Yw);
    else
      y_col<<<dim3(BB*8*16), 256, 0, stream>>>(x, ATTN, Yw);

    // O1[bs, h*32+d] = sum_c Y[bs,h,c] * Wv[c,h*32+d]  (batched over h)
    wmma_gemm<<<dim3(((BS/16)*(DD/16) + 7)/8, HH), 256, 0, stream>>>(
        Yw, Wv, O1, BS, DD, CC,
        HH*CC, 1, HID, 1, HID, 1,
        (long)CC, (long)DD, (long)DD);
    // += mem_kv value path
    mem_add<<<dim3(BS), 256, 0, stream>>>(AMEM, mem_v, O1);

    // AX = O1 @ Wo   (1024x128x256)
    wmma_gemm<<<dim3(((BS/16)*(CC/16) + 7)/8, 1), 256, 0, stream>>>(
        O1, Wo, AX[p], BS, CC, HID, HID, 1, CC, 1, CC, 1, 0, 0, 0);

    // PRJ[p][b,o,n] = sum_c Wr[o, p*C + c] * AX[b,n,c]  (batched over b)
    wmma_gemm<<<dim3(((CC/16)*(SS/16) + 7)/8, BB), 256, 0, stream>>>(
        Wr + p * CC, AX[p], PRJ[p], CC, SS, CC,
        2*CC, 1, 1, CC, SS, 1,
        0L, (long)(SS*CC), (long)(CC*SS));
  }

  // out[b,o,n,m] = PR[b,o,n] + PC[b,o,m]
  final_add<<<dim3(BB*CC*SS), 256, 0, stream>>>(PRJ[0], PRJ[1], out);
}